// SpikingRON_67929202753536
// MI455X (gfx1250) — compile-verified
//
#include <hip/hip_runtime.h>
#include <hip/hip_bf16.h>

// Spiking RON forward for MI455X (gfx1250).
// Persistent-state recurrent kernel:
//   - 8 workgroups x 512 threads (16 waves). Workgroup = 16 batch rows.
//   - Wave w owns hidden-column tile [16w, 16w+16).
//   - h2h / x2h B-fragments live in VGPRs for the whole T loop.
//   - hy is exchanged via double-buffered LDS (1 barrier/step).
//   - x_t is double-buffered: prefetched to regs during step t, stored to LDS
//     for step t+1 before the barrier.
//   - GEMMs use V_WMMA_F32_16X16X4_F32 (full f32 precision, matrix pipe).

typedef __attribute__((ext_vector_type(2))) float v2f;
typedef __attribute__((ext_vector_type(8))) float v8f;

#define BB    128
#define TT    1000
#define NI    64
#define NH    256
#define DTC   0.05f
#define GAMMA 2.7f
#define EPSI  4.7f

#define HY_STR 260   // 256 + 4 pad -> bank-conflict-free A-fragment reads
#define XT_STR 68    // 64 + 4 pad

__global__ __launch_bounds__(512, 1)
void spiking_ron_kernel(const float* __restrict__ x,
                        const float* __restrict__ h2h,
                        const float* __restrict__ x2h,
                        float* __restrict__ out)
{
    __shared__ float hy_sh[2][16 * HY_STR];
    __shared__ float xt_sh[2][16 * XT_STR];

    const int tid  = threadIdx.x;
    const int lane = tid & 31;
    const int wave = tid >> 5;          // 0..15 : hidden column tile
    const int b0   = blockIdx.x * 16;   // batch tile base
    const int n0   = wave * 16;         // hidden col base for this wave

    // ---- Load B-matrix weight fragments into registers (kept for all T) ----
    // B fragment layout for V_WMMA_F32_16X16X4_F32 (4x16 KxN):
    //   VGPR0: lanes 0-15 K=k0,   lanes 16-31 K=k0+2  (N = lane&15)
    //   VGPR1: lanes 0-15 K=k0+1, lanes 16-31 K=k0+3
    const int nn    = n0 + (lane & 15);
    const int khalf = (lane >> 4) * 2;   // 0 (lanes 0-15) or 2 (lanes 16-31)

    v2f wh[64];   // h2h: K=256 -> 64 fragments of K=4
    v2f wx[16];   // x2h: K=64  -> 16 fragments of K=4
#pragma unroll
    for (int f = 0; f < 64; ++f) {
        const int k = 4 * f + khalf;
        wh[f].x = h2h[(k + 0) * NH + nn];
        wh[f].y = h2h[(k + 1) * NH + nn];
    }
#pragma unroll
    for (int f = 0; f < 16; ++f) {
        const int k = 4 * f + khalf;
        wx[f].x = x2h[(k + 0) * NH + nn];
        wx[f].y = x2h[(k + 1) * NH + nn];
    }

    // ---- Init: zero hy buffer 0, preload x_t=0 tile ----
    for (int i = tid; i < 16 * HY_STR; i += 512) hy_sh[0][i] = 0.0f;

    const int xe = tid * 2;           // 1024 floats / 512 threads = 2 each
    const int xr = xe >> 6;           // batch row in tile (0..15)
    const int xc = xe & 63;           // input col (even)
    {
        v2f v = *(const v2f*)(x + ((size_t)(b0 + xr) * TT + 0) * NI + xc);
        *(v2f*)&xt_sh[0][xr * XT_STR + xc] = v;
    }
    __syncthreads();

    // ---- Persistent state (WMMA C layout: VGPR j -> rows j / j+8, col lane&15)
    v8f u  = {};
    v8f hz = {};
    v8f hy = {};

    // A-fragment LDS addressing: lane reads row (lane&15), k = 4f + khalf
    const int arow = lane & 15;
    const int aoff = arow * HY_STR + khalf;
    const int xoff = arow * XT_STR + khalf;

    // Output addressing. d_out = [hys | hzs | us | spikes], each [B,T,H].
    const unsigned BTH   = (unsigned)BB * TT * NH;          // 32,768,000
    const unsigned TH    = (unsigned)TT * NH;
    const unsigned obase = ((unsigned)(b0 + (lane >> 4) * 8)) * TH
                         + (unsigned)n0 + (unsigned)(lane & 15);

    for (int t = 0; t < TT; ++t) {
        const int cur = t & 1;
        const int nxt = cur ^ 1;

        // Prefetch x_{t+1} into registers (latency hidden under the GEMM).
        v2f xpre = {};
        if (t + 1 < TT) {
            xpre = *(const v2f*)(x + ((size_t)(b0 + xr) * TT + (t + 1)) * NI + xc);
        }

        // Spiking layer: threshold + reset (uses u from end of previous step).
        v8f spike;
#pragma unroll
        for (int j = 0; j < 8; ++j) {
            const bool s = u[j] > 1.0f;
            spike[j] = s ? 1.0f : 0.0f;
            u[j]     = s ? 0.0f : u[j];
        }

        // acc = hy_old @ h2h + x_t @ x2h   (f32 WMMA, K=4 per instruction)
        v8f acc = {};
        {
            const float* hA = &hy_sh[cur][aoff];
#pragma unroll
            for (int f = 0; f < 64; ++f) {
                v2f a = *(const v2f*)(hA + 4 * f);
                acc = __builtin_amdgcn_wmma_f32_16x16x4_f32(
                        false, a, false, wh[f], (short)0, acc, false, false);
            }
            const float* xA = &xt_sh[cur][xoff];
#pragma unroll
            for (int f = 0; f < 16; ++f) {
                v2f a = *(const v2f*)(xA + 4 * f);
                acc = __builtin_amdgcn_wmma_f32_16x16x4_f32(
                        false, a, false, wx[f], (short)0, acc, false, false);
            }
        }

        // Pointwise state updates (RC*DT == DTC):
        //   u   = u + (-u + acc)*DTC          (post-reset u)
        //   hz  = hz + DTC*(u_new - GAMMA*hy_old - EPSI*hz)
        //   hy  = hy_old + DTC*hz_new
#pragma unroll
        for (int j = 0; j < 8; ++j) {
            const float un  = u[j] + (acc[j] - u[j]) * DTC;
            const float hzn = hz[j] + DTC * (un - GAMMA * hy[j] - EPSI * hz[j]);
            const float hyn = hy[j] + DTC * hzn;
            u[j] = un; hz[j] = hzn; hy[j] = hyn;
        }

        // Emit outputs for this step (coalesced 64B per half-wave).
        const unsigned ob = obase + (unsigned)t * NH;
#pragma unroll
        for (int j = 0; j < 8; ++j) {
            const unsigned o = ob + (unsigned)j * TH;
            out[o]           = hy[j];
            out[BTH + o]     = hz[j];
            out[2u*BTH + o]  = u[j];
            out[3u*BTH + o]  = spike[j];
        }

        // Publish new hy to the other LDS buffer (row = j + 8*(lane>>4)).
        {
            const int rbase = (lane >> 4) * 8;
            const int col   = n0 + (lane & 15);
#pragma unroll
            for (int j = 0; j < 8; ++j) {
                hy_sh[nxt][(rbase + j) * HY_STR + col] = hy[j];
            }
        }

        // Commit prefetched x_{t+1} tile.
        if (t + 1 < TT) {
            *(v2f*)&xt_sh[nxt][xr * XT_STR + xc] = xpre;
        }

        __syncthreads();
    }
}

extern "C" void kernel_launch(void* const* d_in, const int* in_sizes, int n_in,
                              void* d_out, int out_size, void* d_ws, size_t ws_size,
                              hipStream_t stream) {
    const float* x   = (const float*)d_in[0];   // [128,1000,64]
    const float* h2h = (const float*)d_in[1];   // [256,256]
    const float* x2h = (const float*)d_in[2];   // [64,256]
    float* out = (float*)d_out;                 // 4 x [128,1000,256] concat

    dim3 grid(BB / 16);   // 8 workgroups, one per 16-row batch tile
    dim3 block(512);      // 16 waves of 32
    spiking_ron_kernel<<<grid, block, 0, stream>>>(x, h2h, x2h, out);
}